// RobertaAttRNNHead_29875792511857
// MI455X (gfx1250) — compile-verified
//
#include <hip/hip_runtime.h>
#include <hip/hip_bf16.h>
#include <math.h>

typedef __attribute__((ext_vector_type(2))) float v2f;
typedef __attribute__((ext_vector_type(8))) float v8f;

// ---------------------------------------------------------------------------
// Kernel 1: combined[b, 0:H] = emb[ids[b]]; combined[b, H:2H] = hidden[b]
// ---------------------------------------------------------------------------
__global__ void prep_kernel(const int* __restrict__ ids,
                            const float* __restrict__ emb,
                            const float* __restrict__ hidden,
                            float* __restrict__ comb,
                            int Bn, int Hn) {
    int i = blockIdx.x * blockDim.x + threadIdx.x;
    if (i >= Bn * Hn) return;
    int b = i / Hn, h = i - b * Hn;
    comb[(size_t)b * 2 * Hn + h]      = emb[(size_t)ids[b] * Hn + h];
    comb[(size_t)b * 2 * Hn + Hn + h] = hidden[i];
}

// ---------------------------------------------------------------------------
// Kernel 2: fp32 WMMA GEMM  C[M,N] = A[M,K] @ W[K,N] + bias  (opt ReLU)
// One wave (32 threads) per 32x64 output tile (2x4 grid of 16x16 WMMA tiles),
// V_WMMA_F32_16X16X4_F32, 8 WMMAs per K-step of 4.
// Requires: M % 32 == 0, K % 4 == 0.  N may be ragged.
//
// Ragged-N handling: out-of-range lanes load from a CLAMPED in-bounds column
// (no EXEC masking / cndmask in the loop). Since D[:,n] depends only on
// B[:,n], garbage columns never pollute valid ones; they are simply not
// stored in the epilogue.
// ---------------------------------------------------------------------------
__global__ __launch_bounds__(32) void gemm_f32_wmma(
    const float* __restrict__ A, const float* __restrict__ W,
    const float* __restrict__ bias, float* __restrict__ C,
    int M, int N, int K, int relu)
{
    const int lane = threadIdx.x;      // 0..31
    const int n16  = lane & 15;        // row (A) / col (B,C) within 16
    const int kh   = lane >> 4;        // k-half select: ks 2*kh, 2*kh+1
    const int tilesN = (N + 63) >> 6;
    const int bm = blockIdx.x / tilesN;
    const int bn = blockIdx.x - bm * tilesN;
    const int m0 = bm * 32;
    const int n0 = bn * 64;

    const int r0  = m0 + n16;          // A row, M-tile 0
    const int r1  = m0 + 16 + n16;     // A row, M-tile 1
    const int cA  = 2 * kh;            // k offset within the 4-wide step

    int  cB[4]; bool ok[4]; int cBc[4];
    #pragma unroll
    for (int j = 0; j < 4; ++j) {
        cB[j]  = n0 + 16 * j + n16;    // B col, N-tile j
        ok[j]  = cB[j] < N;
        cBc[j] = ok[j] ? cB[j] : (N - 1);   // clamped (always in bounds)
    }

    // strength-reduced lane base pointers
    const float* Ap0 = A + (size_t)r0 * K + cA;
    const float* Ap1 = A + (size_t)r1 * K + cA;
    const float* Wp0 = W + (size_t)cA * N + cBc[0];
    const float* Wp1 = W + (size_t)cA * N + cBc[1];
    const float* Wp2 = W + (size_t)cA * N + cBc[2];
    const float* Wp3 = W + (size_t)cA * N + cBc[3];
    const size_t wstep = (size_t)4 * N;

    v8f c00 = {}, c01 = {}, c02 = {}, c03 = {};
    v8f c10 = {}, c11 = {}, c12 = {}, c13 = {};

    #pragma unroll 2
    for (int k = 0; k < K; k += 4) {
        v2f a0 = *(const v2f*)Ap0;  Ap0 += 4;     // A[r][k+2kh], A[r][k+2kh+1]
        v2f a1 = *(const v2f*)Ap1;  Ap1 += 4;
        v2f b0, b1, b2, b3;
        b0.x = Wp0[0]; b0.y = Wp0[N];             // W[k+2kh][c], W[k+2kh+1][c]
        b1.x = Wp1[0]; b1.y = Wp1[N];
        b2.x = Wp2[0]; b2.y = Wp2[N];
        b3.x = Wp3[0]; b3.y = Wp3[N];
        Wp0 += wstep; Wp1 += wstep; Wp2 += wstep; Wp3 += wstep;

        c00 = __builtin_amdgcn_wmma_f32_16x16x4_f32(false, a0, false, b0, (short)0, c00, false, false);
        c01 = __builtin_amdgcn_wmma_f32_16x16x4_f32(false, a0, false, b1, (short)0, c01, false, false);
        c02 = __builtin_amdgcn_wmma_f32_16x16x4_f32(false, a0, false, b2, (short)0, c02, false, false);
        c03 = __builtin_amdgcn_wmma_f32_16x16x4_f32(false, a0, false, b3, (short)0, c03, false, false);
        c10 = __builtin_amdgcn_wmma_f32_16x16x4_f32(false, a1, false, b0, (short)0, c10, false, false);
        c11 = __builtin_amdgcn_wmma_f32_16x16x4_f32(false, a1, false, b1, (short)0, c11, false, false);
        c12 = __builtin_amdgcn_wmma_f32_16x16x4_f32(false, a1, false, b2, (short)0, c12, false, false);
        c13 = __builtin_amdgcn_wmma_f32_16x16x4_f32(false, a1, false, b3, (short)0, c13, false, false);
    }

    float bs[4];
    #pragma unroll
    for (int j = 0; j < 4; ++j) bs[j] = (bias != nullptr) ? bias[cBc[j]] : 0.0f;

    v8f acc0[4] = { c00, c01, c02, c03 };   // M-tile 0 (rows m0 .. m0+15)
    v8f acc1[4] = { c10, c11, c12, c13 };   // M-tile 1 (rows m0+16 .. m0+31)

    #pragma unroll
    for (int v = 0; v < 8; ++v) {
        const int mA = m0 + v + 8 * kh;        // rows 0..15 of tile
        const int mB = mA + 16;                // rows 16..31 of tile
        #pragma unroll
        for (int j = 0; j < 4; ++j) {
            if (ok[j]) {
                float x0 = acc0[j][v] + bs[j]; if (relu) x0 = fmaxf(x0, 0.0f);
                C[(size_t)mA * N + cB[j]] = x0;
                float x1 = acc1[j][v] + bs[j]; if (relu) x1 = fmaxf(x1, 0.0f);
                C[(size_t)mB * N + cB[j]] = x1;
            }
        }
    }
}

// ---------------------------------------------------------------------------
// Kernel 3: row softmax over N (in place in ws) + copy to d_out attn region
// blockDim = 256
// ---------------------------------------------------------------------------
__global__ void softmax_rows(float* __restrict__ x, float* __restrict__ out2, int N) {
    const int b = blockIdx.x;
    float* row = x + (size_t)b * N;
    __shared__ float red[256];
    const int t = threadIdx.x;

    float m = -INFINITY;
    for (int i = t; i < N; i += 256) m = fmaxf(m, row[i]);
    red[t] = m; __syncthreads();
    for (int s = 128; s > 0; s >>= 1) { if (t < s) red[t] = fmaxf(red[t], red[t + s]); __syncthreads(); }
    m = red[0]; __syncthreads();

    float sum = 0.0f;
    for (int i = t; i < N; i += 256) sum += expf(row[i] - m);
    red[t] = sum; __syncthreads();
    for (int s = 128; s > 0; s >>= 1) { if (t < s) red[t] += red[t + s]; __syncthreads(); }
    const float inv = 1.0f / red[0]; __syncthreads();

    for (int i = t; i < N; i += 256) {
        float w = expf(row[i] - m) * inv;
        row[i] = w;
        out2[(size_t)b * N + i] = w;
    }
}

// ---------------------------------------------------------------------------
// Kernel 4: attn_applied[b,h] = sum_l w[b,l] * enc[b,l,h]
// Written into comb[b, H:2H] (forms [emb | attn_applied] in place).
// Pure streaming kernel: 512MB of encoder_outputs at float4 granularity.
// blockDim = H/4 (=256)
// ---------------------------------------------------------------------------
__global__ void attn_apply(const float* __restrict__ w,
                           const float* __restrict__ enc,
                           float* __restrict__ comb,
                           int Ln, int Hn) {
    const int b = blockIdx.x;
    const int t = threadIdx.x;            // 0 .. Hn/4-1
    __shared__ float ws_[512];
    for (int i = t; i < Ln; i += blockDim.x) ws_[i] = w[(size_t)b * Ln + i];
    __syncthreads();

    const int hv = Hn >> 2;
    const float4* e = (const float4*)(enc + (size_t)b * Ln * Hn);
    float4 acc = make_float4(0.f, 0.f, 0.f, 0.f);
    for (int l = 0; l < Ln; ++l) {
        const float wv = ws_[l];
        const float4 v = e[(size_t)l * hv + t];
        acc.x += wv * v.x; acc.y += wv * v.y;
        acc.z += wv * v.z; acc.w += wv * v.w;
    }
    float* dst = comb + (size_t)b * 2 * Hn + Hn + t * 4;
    dst[0] = acc.x; dst[1] = acc.y; dst[2] = acc.z; dst[3] = acc.w;
}

// ---------------------------------------------------------------------------
// Kernel 5: GRU gates (PyTorch order: reset, update, new)
// ---------------------------------------------------------------------------
__global__ void gru_elem(const float* __restrict__ gi, const float* __restrict__ gh,
                         const float* __restrict__ h0, float* __restrict__ newh,
                         int Bn, int Hn) {
    int i = blockIdx.x * blockDim.x + threadIdx.x;
    if (i >= Bn * Hn) return;
    int b = i / Hn, h = i - b * Hn;
    const float* gib = gi + (size_t)b * 3 * Hn;
    const float* ghb = gh + (size_t)b * 3 * Hn;
    float r = 1.0f / (1.0f + expf(-(gib[h] + ghb[h])));
    float z = 1.0f / (1.0f + expf(-(gib[Hn + h] + ghb[Hn + h])));
    float n = tanhf(gib[2 * Hn + h] + r * ghb[2 * Hn + h]);
    newh[i] = (1.0f - z) * n + z * h0[i];
}

// ---------------------------------------------------------------------------
// Kernel 6: in-place row log_softmax over N.  blockDim = 256
// ---------------------------------------------------------------------------
__global__ void log_softmax_rows(float* __restrict__ x, int N) {
    const int b = blockIdx.x;
    float* row = x + (size_t)b * N;
    __shared__ float red[256];
    const int t = threadIdx.x;

    float m = -INFINITY;
    for (int i = t; i < N; i += 256) m = fmaxf(m, row[i]);
    red[t] = m; __syncthreads();
    for (int s = 128; s > 0; s >>= 1) { if (t < s) red[t] = fmaxf(red[t], red[t + s]); __syncthreads(); }
    m = red[0]; __syncthreads();

    float sum = 0.0f;
    for (int i = t; i < N; i += 256) sum += expf(row[i] - m);
    red[t] = sum; __syncthreads();
    for (int s = 128; s > 0; s >>= 1) { if (t < s) red[t] += red[t + s]; __syncthreads(); }
    const float lse = m + logf(red[0]); __syncthreads();

    for (int i = t; i < N; i += 256) row[i] -= lse;
}

// ---------------------------------------------------------------------------
extern "C" void kernel_launch(void* const* d_in, const int* in_sizes, int n_in,
                              void* d_out, int out_size, void* d_ws, size_t ws_size,
                              hipStream_t stream) {
    const int*   ids    = (const int*)  d_in[0];
    const float* hidden = (const float*)d_in[1];
    const float* enc    = (const float*)d_in[2];
    const float* emb    = (const float*)d_in[3];
    const float* attn_w = (const float*)d_in[4];
    const float* attn_b = (const float*)d_in[5];
    const float* comb_w = (const float*)d_in[6];
    const float* comb_b = (const float*)d_in[7];
    const float* w_ih   = (const float*)d_in[8];
    const float* w_hh   = (const float*)d_in[9];
    const float* b_ih   = (const float*)d_in[10];
    const float* b_hh   = (const float*)d_in[11];
    const float* out_w  = (const float*)d_in[12];
    const float* out_b  = (const float*)d_in[13];

    const int Bn = in_sizes[0];                       // 512
    const int Hn = in_sizes[1] / Bn;                  // 1024
    const int Ln = in_sizes[2] / (Bn * Hn);           // 256
    const int Vn = in_sizes[13];                      // 10000

    // workspace layout (floats)
    float* w    = (float*)d_ws;
    float* comb = w;                                  // B * 2H  ([emb|hidden] -> [emb|attn_applied])
    float* attn = comb + (size_t)Bn * 2 * Hn;         // B * L
    float* xbuf = attn + (size_t)Bn * Ln;             // B * H
    float* gi   = xbuf + (size_t)Bn * Hn;             // B * 3H
    float* gh   = gi   + (size_t)Bn * 3 * Hn;         // B * 3H

    // output layout: [out (B*V) | new_hidden (B*H) | attn_weights (B*L)]
    float* out      = (float*)d_out;
    float* newh     = out + (size_t)Bn * Vn;
    float* attn_out = newh + (size_t)Bn * Hn;

    auto tiles = [](int M, int N) { return (M / 32) * ((N + 63) / 64); };

    // 1. gather + concat
    prep_kernel<<<(Bn * Hn + 255) / 256, 256, 0, stream>>>(ids, emb, hidden, comb, Bn, Hn);
    // 2. attention logits: (B,2H)@(2H,L)+b
    gemm_f32_wmma<<<tiles(Bn, Ln), 32, 0, stream>>>(comb, attn_w, attn_b, attn, Bn, Ln, 2 * Hn, 0);
    // 3. softmax rows -> ws + output region
    softmax_rows<<<Bn, 256, 0, stream>>>(attn, attn_out, Ln);
    // 4. weighted sum over encoder outputs -> comb[:, H:2H]
    attn_apply<<<Bn, Hn / 4, 0, stream>>>(attn, enc, comb, Ln, Hn);
    // 5. x = relu([emb|attn_applied]@comb_w + b)
    gemm_f32_wmma<<<tiles(Bn, Hn), 32, 0, stream>>>(comb, comb_w, comb_b, xbuf, Bn, Hn, 2 * Hn, 1);
    // 6. GRU input/hidden GEMMs
    gemm_f32_wmma<<<tiles(Bn, 3 * Hn), 32, 0, stream>>>(xbuf, w_ih, b_ih, gi, Bn, 3 * Hn, Hn, 0);
    gemm_f32_wmma<<<tiles(Bn, 3 * Hn), 32, 0, stream>>>(hidden, w_hh, b_hh, gh, Bn, 3 * Hn, Hn, 0);
    // 7. GRU gates -> new_hidden (directly into output region)
    gru_elem<<<(Bn * Hn + 255) / 256, 256, 0, stream>>>(gi, gh, hidden, newh, Bn, Hn);
    // 8. logits = new_hidden @ out_w + out_b  (into d_out), then log_softmax in place
    gemm_f32_wmma<<<tiles(Bn, Vn), 32, 0, stream>>>(newh, out_w, out_b, out, Bn, Vn, Hn, 0);
    log_softmax_rows<<<Bn, 256, 0, stream>>>(out, Vn);
}